// ModelNew_7868380086952
// MI455X (gfx1250) — compile-verified
//
#include <hip/hip_runtime.h>
#include <stdint.h>

// MoE combine: out[token_idx[r]] += gates[r] * expert_hidden[r]
// rows=32768, hidden=4096, tokens=16384, all fp32.
// Bandwidth-bound (~1 GB traffic); use CDNA5 async-to-LDS staging + native
// fp32 global atomics.

#define HIDDEN      4096
#define NUM_ROWS    32768
#define NUM_TOKENS  16384
#define BLOCK       256                 // 8 wave32s
#define WAVES       (BLOCK / 32)
#define WAVE_FLOATS (HIDDEN / WAVES)    // 512 floats = 2 KB per wave per row
#define LOADS_PER_WAVE 4                // 4 x (32 lanes x 16B) = 2 KB

__global__ __launch_bounds__(BLOCK) void moe_zero_out_kernel(float4* __restrict__ out4, int n4) {
    int i = blockIdx.x * blockDim.x + threadIdx.x;
    const int stride = gridDim.x * blockDim.x;
    const float4 z = make_float4(0.f, 0.f, 0.f, 0.f);
    for (; i < n4; i += stride) out4[i] = z;   // global_store_b128; warms L2 for atomics
}

__device__ __forceinline__ void stage_row_async(const float* __restrict__ hidden,
                                                int row, float* lds_buf,
                                                int wave, int lane) {
    // Each wave stages its own 2 KB slice: 4 async b128 instructions,
    // each instruction = 32 lanes x 16 B contiguous (512 B coalesced).
    const int base = wave * WAVE_FLOATS + lane * 4;
    const float* gbase = hidden + (size_t)row * HIDDEN + base;
#pragma unroll
    for (int i = 0; i < LOADS_PER_WAVE; ++i) {
        const unsigned long long gaddr =
            (unsigned long long)(uintptr_t)(gbase + i * 128);          // 128 floats = 512 B
        const unsigned lds_addr =
            (unsigned)(uintptr_t)(lds_buf + base + i * 128);           // low 32b = LDS offset
        asm volatile("global_load_async_to_lds_b128 %0, %1, off th:TH_LOAD_NT"
                     :: "v"(lds_addr), "v"(gaddr)
                     : "memory");
    }
}

__global__ __launch_bounds__(BLOCK) void moe_combine_scatter_kernel(
        const float* __restrict__ expert_hidden,
        const int*   __restrict__ token_idx,
        const float* __restrict__ gates,
        float*       __restrict__ out) {
    __shared__ float buf[2][HIDDEN];   // 32 KB double buffer, partitioned per-wave

    const int wave = threadIdx.x >> 5;
    const int lane = threadIdx.x & 31;
    const int row0 = blockIdx.x;
    const int rstride = gridDim.x;

    // Prologue: stage first row into buffer 0.
    stage_row_async(expert_hidden, row0, buf[0], wave, lane);

    int cur = 0;
    for (int row = row0; row < NUM_ROWS; row += rstride) {
        const int nxt = row + rstride;
        if (nxt < NUM_ROWS) {
            // Prefetch next row into the other buffer, then wait for current
            // buffer only (<= 4 outstanding async instructions on this wave).
            stage_row_async(expert_hidden, nxt, buf[cur ^ 1], wave, lane);
            asm volatile("s_wait_asynccnt 0x4" ::: "memory");
        } else {
            asm volatile("s_wait_asynccnt 0x0" ::: "memory");
        }

        const int   tok = token_idx[row];   // uniform -> scalar load
        const float g   = gates[row];
        float* orow = out + (size_t)tok * HIDDEN;

        // lane l handles columns wave*512 + k*32 + l: every atomic instruction
        // covers 32 consecutive fp32 = one 128 B cacheline at L2 (coalesced RMW).
        const int cbase = wave * WAVE_FLOATS + lane;
#pragma unroll
        for (int k = 0; k < 16; ++k) {
            const int c = cbase + k * 32;
            const float v = buf[cur][c] * g;
            const unsigned long long a = (unsigned long long)(uintptr_t)(orow + c);
            asm volatile("global_atomic_add_f32 %0, %1, off scope:SCOPE_DEV"
                         :: "v"(a), "v"(v)
                         : "memory");
        }
        cur ^= 1;
    }
}

extern "C" void kernel_launch(void* const* d_in, const int* in_sizes, int n_in,
                              void* d_out, int out_size, void* d_ws, size_t ws_size,
                              hipStream_t stream) {
    (void)in_sizes; (void)n_in; (void)d_ws; (void)ws_size; (void)out_size;

    const float* expert_hidden = (const float*)d_in[0];
    const int*   token_idx     = (const int*)d_in[1];
    const float* gates         = (const float*)d_in[2];
    float*       out           = (float*)d_out;

    // 1) Zero the 256 MB output (d_out is poisoned; fp atomics need a clean base).
    const int n4 = (NUM_TOKENS * HIDDEN) / 4;   // 16,777,216 float4
    moe_zero_out_kernel<<<8192, BLOCK, 0, stream>>>((float4*)out, n4);

    // 2) Async-staged, double-buffered scatter-add. 2048 blocks x 16 rows each.
    moe_combine_scatter_kernel<<<2048, BLOCK, 0, stream>>>(
        expert_hidden, token_idx, gates, out);
}